// CharEmbeddingV411_4063039062449
// MI455X (gfx1250) — compile-verified
//
#include <hip/hip_runtime.h>
#include <hip/hip_bf16.h>
#include <math.h>

// ---------------------------------------------------------------------------
// CharEmbeddingV411 for MI455X (gfx1250, wave32)
//
// The whole network is a pure function of the token id (0..255):
//   out[b,s,:] = F(tokens[b,s])
// Kernel 1: build the 256x11 table F (WMMA f32 16x16x4 for the 73->11 GEMM).
// Kernel 2: memory-bound gather, staged through LDS for coalesced b128 stores.
// ---------------------------------------------------------------------------

typedef __attribute__((ext_vector_type(2))) float v2f;
typedef __attribute__((ext_vector_type(8))) float v8f;

#define KPAD 80   // 73 features padded to multiple of 4 (K for WMMA f32 16x16x4)
#define NPAD 16   // 11 outputs padded to 16

__device__ __forceinline__ float swishf(float z) {
    return z / (1.0f + __expf(-z));
}

// One workgroup of 256 threads (8 wave32 waves). Thread t == token id t.
__global__ __launch_bounds__(256) void build_table_kernel(
    const float* __restrict__ emb,
    const float* __restrict__ W112, const float* __restrict__ b112,
    const float* __restrict__ W113, const float* __restrict__ b113,
    const float* __restrict__ W115, const float* __restrict__ b115,
    const float* __restrict__ W117, const float* __restrict__ b117,
    const float* __restrict__ W21,  const float* __restrict__ b21,
    const float* __restrict__ W31,  const float* __restrict__ b31,
    const float* __restrict__ W51,  const float* __restrict__ b51,
    const float* __restrict__ W71,  const float* __restrict__ b71,
    const float* __restrict__ W111, const float* __restrict__ b111,
    const float* __restrict__ Wout, const float* __restrict__ bout,
    float* __restrict__ table,      // [256*11]   output lookup table
    float* __restrict__ featG,      // [256*KPAD] scratch: feature matrix (A)
    float* __restrict__ WbG)        // [KPAD*NPAD] scratch: padded Wout (B)
{
    const int t = threadIdx.x; // token id 0..255

    // ---------- phase A: scalar fp32 feature computation (exact) ----------
    float x[11];
#pragma unroll
    for (int i = 0; i < 11; ++i) x[i] = emb[t * 11 + i];

    float d112[2], d113[3], d115[5], d117[7];
#pragma unroll
    for (int j = 0; j < 2; ++j) {
        float s = b112[j];
#pragma unroll
        for (int i = 0; i < 11; ++i) s += x[i] * W112[i * 2 + j];
        d112[j] = swishf(s);
    }
#pragma unroll
    for (int j = 0; j < 3; ++j) {
        float s = b113[j];
#pragma unroll
        for (int i = 0; i < 11; ++i) s += x[i] * W113[i * 3 + j];
        d113[j] = swishf(s);
    }
#pragma unroll
    for (int j = 0; j < 5; ++j) {
        float s = b115[j];
#pragma unroll
        for (int i = 0; i < 11; ++i) s += x[i] * W115[i * 5 + j];
        d115[j] = swishf(s);
    }
#pragma unroll
    for (int j = 0; j < 7; ++j) {
        float s = b117[j];
#pragma unroll
        for (int i = 0; i < 11; ++i) s += x[i] * W117[i * 7 + j];
        d117[j] = swishf(s);
    }

    float s21 = b21[0];
#pragma unroll
    for (int i = 0; i < 2; ++i) s21 += d112[i] * W21[i];
    float s31 = b31[0];
#pragma unroll
    for (int i = 0; i < 3; ++i) s31 += d113[i] * W31[i];
    float s51 = b51[0];
#pragma unroll
    for (int i = 0; i < 5; ++i) s51 += d115[i] * W51[i];
    float s71 = b71[0];
#pragma unroll
    for (int i = 0; i < 7; ++i) s71 += d117[i] * W71[i];
    float s111 = b111[0];
#pragma unroll
    for (int i = 0; i < 11; ++i) s111 += x[i] * W111[i];

    float sg[5];
    sg[0] = swishf(s21);
    sg[1] = swishf(s31);
    sg[2] = swishf(s51);
    sg[3] = swishf(s71);
    sg[4] = swishf(s111);

    const int PI[10] = {0, 0, 0, 0, 1, 1, 1, 2, 2, 3};
    const int PJ[10] = {1, 2, 3, 4, 2, 3, 4, 3, 4, 4};

    // concat order: x(11), d112(2), d113(3), d115(5), d117(7),
    //               singles(5), adds(10), muls(10), mins(10), maxs(10) = 73
    float f[KPAD];
    int c = 0;
#pragma unroll
    for (int i = 0; i < 11; ++i) f[c++] = x[i];
#pragma unroll
    for (int i = 0; i < 2; ++i) f[c++] = d112[i];
#pragma unroll
    for (int i = 0; i < 3; ++i) f[c++] = d113[i];
#pragma unroll
    for (int i = 0; i < 5; ++i) f[c++] = d115[i];
#pragma unroll
    for (int i = 0; i < 7; ++i) f[c++] = d117[i];
#pragma unroll
    for (int i = 0; i < 5; ++i) f[c++] = sg[i];
#pragma unroll
    for (int k = 0; k < 10; ++k) f[c++] = sg[PI[k]] + sg[PJ[k]];
#pragma unroll
    for (int k = 0; k < 10; ++k) f[c++] = sg[PI[k]] * sg[PJ[k]];
#pragma unroll
    for (int k = 0; k < 10; ++k) f[c++] = fminf(sg[PI[k]], sg[PJ[k]]);
#pragma unroll
    for (int k = 0; k < 10; ++k) f[c++] = fmaxf(sg[PI[k]], sg[PJ[k]]);
#pragma unroll
    for (int k = 73; k < KPAD; ++k) f[k] = 0.0f;

#pragma unroll
    for (int k = 0; k < KPAD; ++k) featG[t * KPAD + k] = f[k];

    // stage padded B matrix (Wout is [73][11] row-major -> [KPAD][NPAD])
    for (int i = t; i < KPAD * NPAD; i += 256) {
        const int k = i / NPAD, n = i % NPAD;
        WbG[i] = (k < 73 && n < 11) ? Wout[k * 11 + n] : 0.0f;
    }

    __threadfence_block();
    __syncthreads();

    // ---------- phase B: 256x80 @ 80x16 via V_WMMA_F32_16X16X4_F32 ----------
    // A 16x4 fp32 layout: lanes 0-15 (M=row): v0=K0,v1=K1; lanes 16-31: v0=K2,v1=K3
    // B 4x16  fp32 layout: lanes 0-15 (N=col): v0=K0,v1=K1; lanes 16-31: v0=K2,v1=K3
    const int wave = t >> 5;
    const int lane = t & 31;
    const int nl   = lane & 15;
    const int half = (lane < 16) ? 0 : 2;

    const float* frow0 = featG + (wave * 32 + nl) * KPAD;        // tile 0: rows 32w..32w+15
    const float* frow1 = featG + (wave * 32 + 16 + nl) * KPAD;   // tile 1: rows 32w+16..32w+31

    v8f acc0 = {};
    v8f acc1 = {};
#pragma unroll
    for (int k0 = 0; k0 < KPAD; k0 += 4) {
        v2f b;
        b[0] = WbG[(k0 + half) * NPAD + nl];
        b[1] = WbG[(k0 + half + 1) * NPAD + nl];
        v2f a0;
        a0[0] = frow0[k0 + half];
        a0[1] = frow0[k0 + half + 1];
        v2f a1;
        a1[0] = frow1[k0 + half];
        a1[1] = frow1[k0 + half + 1];
        acc0 = __builtin_amdgcn_wmma_f32_16x16x4_f32(false, a0, false, b,
                                                     (short)0, acc0, false, false);
        acc1 = __builtin_amdgcn_wmma_f32_16x16x4_f32(false, a1, false, b,
                                                     (short)0, acc1, false, false);
    }

    // D layout: VGPR v -> M = v (lanes 0-15) / v+8 (lanes 16-31), N = lane%16
    if (nl < 11) {
        const float bo = bout[nl];
        const int mofs = (lane < 16) ? 0 : 8;
        const int m0 = wave * 32 + mofs;
        const int m1 = wave * 32 + 16 + mofs;
#pragma unroll
        for (int v = 0; v < 8; ++v) {
            table[(m0 + v) * 11 + nl] = tanhf(acc0[v] + bo);
            table[(m1 + v) * 11 + nl] = tanhf(acc1[v] + bo);
        }
    }
}

// ---------------------------------------------------------------------------
// Gather kernel: the HBM-bound bulk. Each 256-thread block handles 256 tokens:
// table (11 KB) -> LDS, per-token gather into LDS staging (stride 11 floats is
// conflict-free across 64 banks), then fully coalesced float4 stores of the
// block's contiguous 11264-byte output chunk.
// ---------------------------------------------------------------------------
__global__ __launch_bounds__(256) void gather_kernel(
    const int* __restrict__ tokens,
    const float* __restrict__ table,   // [256*11]
    float* __restrict__ out)           // [B*S*11]
{
    __shared__ float4 tb4[704];   // 256*11 floats = 704 float4
    __shared__ float4 ob4[704];
    float* tb = (float*)tb4;
    float* ob = (float*)ob4;

    const int tid = threadIdx.x;
    const long long blk = blockIdx.x;

    const float4* t4 = (const float4*)table;
#pragma unroll
    for (int i = 0; i < 3; ++i) {
        const int idx = tid + i * 256;
        if (idx < 704) tb4[idx] = t4[idx];
    }
    __syncthreads();

    const int tok = tokens[blk * 256 + tid];
    const float* src = tb + tok * 11;
    float* dst = ob + tid * 11;
#pragma unroll
    for (int j = 0; j < 11; ++j) dst[j] = src[j];
    __syncthreads();

    float4* o4 = ((float4*)out) + blk * 704;
#pragma unroll
    for (int i = 0; i < 3; ++i) {
        const int idx = tid + i * 256;
        if (idx < 704) o4[idx] = ob4[idx];
    }
}

extern "C" void kernel_launch(void* const* d_in, const int* in_sizes, int n_in,
                              void* d_out, int out_size, void* d_ws, size_t ws_size,
                              hipStream_t stream) {
    const int*   tokens = (const int*)  d_in[0];
    const float* emb    = (const float*)d_in[1];
    const float* W112   = (const float*)d_in[2];
    const float* b112   = (const float*)d_in[3];
    const float* W113   = (const float*)d_in[4];
    const float* b113   = (const float*)d_in[5];
    const float* W115   = (const float*)d_in[6];
    const float* b115   = (const float*)d_in[7];
    const float* W117   = (const float*)d_in[8];
    const float* b117   = (const float*)d_in[9];
    const float* W21    = (const float*)d_in[10];
    const float* b21    = (const float*)d_in[11];
    const float* W31    = (const float*)d_in[12];
    const float* b31    = (const float*)d_in[13];
    const float* W51    = (const float*)d_in[14];
    const float* b51    = (const float*)d_in[15];
    const float* W71    = (const float*)d_in[16];
    const float* b71    = (const float*)d_in[17];
    const float* W111   = (const float*)d_in[18];
    const float* b111   = (const float*)d_in[19];
    const float* Wout   = (const float*)d_in[20];
    const float* bout   = (const float*)d_in[21];

    float* out = (float*)d_out;

    // workspace layout (floats): table[4096] (2816 used) | featG[256*KPAD] | WbG[KPAD*NPAD]
    float* ws    = (float*)d_ws;
    float* table = ws;
    float* featG = ws + 4096;
    float* WbG   = ws + 4096 + 256 * KPAD;

    build_table_kernel<<<1, 256, 0, stream>>>(
        emb, W112, b112, W113, b113, W115, b115, W117, b117,
        W21, b21, W31, b31, W51, b51, W71, b71, W111, b111,
        Wout, bout, table, featG, WbG);

    const int n_tokens = in_sizes[0];            // B*S = 1048576
    const int n_blocks = n_tokens / 256;         // 4096
    gather_kernel<<<n_blocks, 256, 0, stream>>>(tokens, table, out);
}